// AttentionRouter_2482491097252
// MI455X (gfx1250) — compile-verified
//
#include <hip/hip_runtime.h>
#include <hip/hip_bf16.h>
#include <math.h>

#define B_ 32
#define S_ 2048
#define H_ 32
#define D_ 128

typedef __attribute__((ext_vector_type(2))) float v2f;
typedef __attribute__((ext_vector_type(8))) float v8f;

// ---------------------------------------------------------------------------
// Kernel 1: segment means over seq for ctx span and q span, x = 0.5*(ctx+q).
// One block per (b,h). 256 threads = 32 lanes (each owns 4 consecutive d via
// float4) x 8 seq slices. LDS reduce across slices (slice 0 sums the other 7).
// ---------------------------------------------------------------------------
__global__ __launch_bounds__(256)
void seg_mean_kernel(const float* __restrict__ pooled,
                     const int* __restrict__ range_ids,
                     float* __restrict__ xout /* [B*H, D] */) {
    __shared__ float4 sdata[256];

    const int bh   = blockIdx.x;          // 0..1023
    const int b    = bh / H_;
    const int h    = bh % H_;
    const int tid  = threadIdx.x;
    const int lane = tid & 31;            // d-group
    const int slc  = tid >> 5;            // 0..7 seq slice
    const int d0   = lane * 4;

    const int s0 = range_ids[b * 6 + 0];
    const int e0 = range_ids[b * 6 + 1];
    const int s1 = range_ids[b * 6 + 2];
    const int e1 = range_ids[b * 6 + 3];

    // base index for (b, s=0, h, d0); stride between seq positions = H_*D_
    const size_t base   = ((size_t)b * S_ * H_ + (size_t)h) * D_ + (size_t)d0;
    const size_t sstride = (size_t)H_ * D_;

    float4 meanC = make_float4(0.f, 0.f, 0.f, 0.f);
    float4 meanQ = make_float4(0.f, 0.f, 0.f, 0.f);

    for (int span = 0; span < 2; ++span) {
        const int ss = (span == 0) ? s0 : s1;
        const int ee = (span == 0) ? e0 : e1;

        float4 acc = make_float4(0.f, 0.f, 0.f, 0.f);
        for (int s = ss + slc; s <= ee; s += 8) {
            const float4 v = *reinterpret_cast<const float4*>(pooled + base + (size_t)s * sstride);
            acc.x += v.x; acc.y += v.y; acc.z += v.z; acc.w += v.w;
        }
        sdata[tid] = acc;
        __syncthreads();

        if (slc == 0) {
            for (int j = 1; j < 8; ++j) {
                const float4 o = sdata[lane + j * 32];
                acc.x += o.x; acc.y += o.y; acc.z += o.z; acc.w += o.w;
            }
            const int cnt = (ee >= ss) ? (ee - ss + 1) : 0;
            const float inv = (cnt > 0) ? (1.0f / (float)cnt) : 0.0f;
            acc.x *= inv; acc.y *= inv; acc.z *= inv; acc.w *= inv;
            if (span == 0) meanC = acc; else meanQ = acc;
        }
        __syncthreads();
    }

    if (slc == 0) {
        float4 x;
        x.x = 0.5f * (meanC.x + meanQ.x);
        x.y = 0.5f * (meanC.y + meanQ.y);
        x.z = 0.5f * (meanC.z + meanQ.z);
        x.w = 0.5f * (meanC.w + meanQ.w);
        *reinterpret_cast<float4*>(xout + (size_t)bh * D_ + d0) = x;
    }
}

// ---------------------------------------------------------------------------
// Kernel 2: fused MLP router with V_WMMA_F32_16X16X4_F32.
// Grid = 64 blocks (16 rows each), 256 threads = 8 waves.
//   Stage 1: X[16x128] @ W1[128x256] -> h1[16x256] (+b1, SiLU) in LDS
//   Stage 2: h1 @ W2[256x128]        -> h2[16x128] (+b2, SiLU) in LDS
//   Stage 3: h2 @ W3[128x2] + b3 -> logits; sigmoid(binary*1.5); outputs
// ---------------------------------------------------------------------------
#define LDS_H1_STRIDE 260   // 256 + 4 pad -> conflict-free column reads
#define LDS_H2_STRIDE 132   // 128 + 4 pad

__device__ __forceinline__ float silu_f(float v) {
    return v / (1.0f + __expf(-v));
}

__global__ __launch_bounds__(256)
void router_mlp_kernel(const float* __restrict__ X /* [1024,128] */,
                       const float* __restrict__ W1 /* [128,256] */,
                       const float* __restrict__ b1 /* [256] */,
                       const float* __restrict__ W2 /* [256,128] */,
                       const float* __restrict__ b2 /* [128] */,
                       const float* __restrict__ W3 /* [128,2] */,
                       const float* __restrict__ b3 /* [2] */,
                       float* __restrict__ out /* 5120 floats */) {
    __shared__ float h1[16 * LDS_H1_STRIDE];
    __shared__ float h2[16 * LDS_H2_STRIDE];

    const int tid  = threadIdx.x;
    const int wave = tid >> 5;     // 0..7
    const int lane = tid & 31;
    const int m0   = blockIdx.x * 16;

    const int lm   = lane & 15;          // row (A) / col (B,C)
    const int khi  = (lane >> 4) * 2;    // K sub-offset within 4-wide K step
    const int crow = (lane >> 4) * 8;    // C-fragment row offset

    // -------- Stage 1: X @ W1 -> h1 (two 16-col tiles per wave) -----------
    for (int t = 0; t < 2; ++t) {
        const int n0 = (wave * 2 + t) * 16;
        v8f c = {};
        for (int k0 = 0; k0 < 128; k0 += 4) {
            const int k = k0 + khi;
            v2f a, bfrag;
            {
                const float2 av = *reinterpret_cast<const float2*>(
                    X + (size_t)(m0 + lm) * 128 + k);
                a.x = av.x; a.y = av.y;
            }
            bfrag.x = W1[(size_t)(k + 0) * 256 + n0 + lm];
            bfrag.y = W1[(size_t)(k + 1) * 256 + n0 + lm];
            c = __builtin_amdgcn_wmma_f32_16x16x4_f32(
                    false, a, false, bfrag, (short)0, c, false, false);
        }
        const float bias = b1[n0 + lm];
        #pragma unroll
        for (int r = 0; r < 8; ++r) {
            const int row = r + crow;
            h1[row * LDS_H1_STRIDE + n0 + lm] = silu_f(c[r] + bias);
        }
    }
    __syncthreads();

    // -------- Stage 2: h1 @ W2 -> h2 (one 16-col tile per wave) -----------
    {
        const int n0 = wave * 16;
        v8f c = {};
        for (int k0 = 0; k0 < 256; k0 += 4) {
            const int k = k0 + khi;
            v2f a, bfrag;
            a.x = h1[lm * LDS_H1_STRIDE + k + 0];
            a.y = h1[lm * LDS_H1_STRIDE + k + 1];
            bfrag.x = W2[(size_t)(k + 0) * 128 + n0 + lm];
            bfrag.y = W2[(size_t)(k + 1) * 128 + n0 + lm];
            c = __builtin_amdgcn_wmma_f32_16x16x4_f32(
                    false, a, false, bfrag, (short)0, c, false, false);
        }
        const float bias = b2[n0 + lm];
        #pragma unroll
        for (int r = 0; r < 8; ++r) {
            const int row = r + crow;
            h2[row * LDS_H2_STRIDE + n0 + lm] = silu_f(c[r] + bias);
        }
    }
    __syncthreads();

    // -------- Stage 3: h2 @ W3 + b3, sigmoid / threshold, write outputs ---
    if (tid < 16) {
        const int row = tid;
        const int m = m0 + row;
        float l0 = b3[0], l1 = b3[1];
        for (int k = 0; k < 128; ++k) {
            const float v = h2[row * LDS_H2_STRIDE + k];
            l0 += v * W3[k * 2 + 0];
            l1 += v * W3[k * 2 + 1];
        }
        const float binary = l1 - l0;
        const float z = 1.0f / (1.0f + __expf(-binary * 1.5f)); // tau = 2/3
        const float hard = (z > 0.5f) ? 1.0f : 0.0f;
        out[m]            = z;       // decisions (z_soft)
        out[1024 + m]     = hard;    // z_hard
        out[2048 + m]     = hard;    // sparse_mask
        out[3072 + 2 * m] = l0;      // logits[...,0]
        out[3072 + 2*m+1] = l1;      // logits[...,1]
    }
}

// ---------------------------------------------------------------------------
extern "C" void kernel_launch(void* const* d_in, const int* in_sizes, int n_in,
                              void* d_out, int out_size, void* d_ws, size_t ws_size,
                              hipStream_t stream) {
    const float* pooled    = (const float*)d_in[0];
    const int*   range_ids = (const int*)d_in[1];
    const float* W1        = (const float*)d_in[2];
    const float* b1        = (const float*)d_in[3];
    const float* W2        = (const float*)d_in[4];
    const float* b2        = (const float*)d_in[5];
    const float* W3        = (const float*)d_in[6];
    const float* b3        = (const float*)d_in[7];
    float* out = (float*)d_out;
    float* X   = (float*)d_ws;   // [1024 x 128] pooled features

    seg_mean_kernel<<<dim3(B_ * H_), dim3(256), 0, stream>>>(pooled, range_ids, X);
    router_mlp_kernel<<<dim3((B_ * H_) / 16), dim3(256), 0, stream>>>(
        X, W1, b1, W2, b2, W3, b3, out);
}